// IPOTSolver_31078383354441
// MI455X (gfx1250) — compile-verified
//
#include <hip/hip_runtime.h>
#include <hip/hip_bf16.h>
#include <stddef.h>

// ---------------------------------------------------------------------------
// IPOT solver for MI455X (gfx1250).
//   K_mat = exp(2*C), C: 32768 x 512 f32.
//   20 iterations of: rowsum = K@sigma ; r = a/(rowsum+eps) ;
//                     colsum = sigma .* (K^T @ r) ; sigma = b/(colsum+eps)
//   output = n * r_i * K[i,j] * sigma_j  (sigma after 19 updates, r from pass 20)
// Rowsums computed with V_WMMA_F32_16X16X4_F32 (f32 matrix pipe, wave32).
// All reductions are atomic-free and order-fixed => bitwise deterministic.
// ---------------------------------------------------------------------------

typedef __attribute__((ext_vector_type(2))) float v2f;
typedef __attribute__((ext_vector_type(4))) float v4f;
typedef __attribute__((ext_vector_type(8))) float v8f;

#define IPOT_N        32768
#define IPOT_K        512
#define IPOT_EPS      1e-12f
#define ROWS_PER_WAVE 16
#define WAVES_PER_WG  4
#define WG_SIZE       (WAVES_PER_WG * 32)            // 128 threads = 4 wave32
#define NUM_TILES     (IPOT_N / ROWS_PER_WAVE)       // 2048
#define GRID_MAIN     (NUM_TILES / WAVES_PER_WG)     // 512 blocks

static_assert(GRID_MAIN == 512, "reduction kernel assumes 512 block partials");

// ---------------- exp materialization (accurate expf, one-time) ------------
__global__ __launch_bounds__(256) void ipot_expm(const float* __restrict__ C,
                                                 float* __restrict__ Km) {
  const size_t total4 = (size_t)IPOT_N * IPOT_K / 4;
  size_t i = (size_t)blockIdx.x * blockDim.x + threadIdx.x;
  if (i >= total4) return;
  const v4f v = ((const v4f*)C)[i];
  v4f w;
  w.x = expf(2.0f * v.x);
  w.y = expf(2.0f * v.y);
  w.z = expf(2.0f * v.z);
  w.w = expf(2.0f * v.w);
  ((v4f*)Km)[i] = w;
}

__global__ void ipot_seed(float* __restrict__ sigma) {
  sigma[threadIdx.x] = 1.0f / (float)IPOT_K;         // <<<1,512>>>
}

__device__ __forceinline__ v4f ipot_exp4(v4f v) {
  v4f w;
  w.x = __expf(2.0f * v.x);
  w.y = __expf(2.0f * v.y);
  w.z = __expf(2.0f * v.z);
  w.w = __expf(2.0f * v.w);
  return w;
}

// ---------------- fused pass: rowsum (WMMA) -> r -> colsum / output --------
template <bool EXPMODE, bool FINAL>
__global__ __launch_bounds__(WG_SIZE)
void ipot_pass(const float* __restrict__ Ksrc,
               const float* __restrict__ sigma,
               float* __restrict__ partial,   // [GRID_MAIN][512]; unused if FINAL
               float* __restrict__ out)       // used only if FINAL
{
  __shared__ float sigma_s[IPOT_K];
  __shared__ float ps[WAVES_PER_WG][IPOT_K];          // per-wave colsum partials
  __shared__ float r_s[WAVES_PER_WG][ROWS_PER_WAVE];

  const int t = threadIdx.x;
  for (int i = t; i < IPOT_K; i += WG_SIZE) sigma_s[i] = sigma[i];
  __syncthreads();

  const int wave = t >> 5;
  const int lane = t & 31;
  const int tile = blockIdx.x * WAVES_PER_WG + wave;  // 16-row tile per wave
  const float* __restrict__ rowbase =
      Ksrc + (size_t)tile * ROWS_PER_WAVE * IPOT_K;

  // ---- Phase A: 16 rowsums via V_WMMA_F32_16X16X4_F32 ----
  // A fragment (16x4 f32): lanes 0-15 rows, VGPR0/1 = K+0/K+1; lanes 16-31 = K+2/K+3.
  // B fragment: sigma replicated across all 16 columns -> every C column = rowsum.
  const int m    = lane & 15;
  const int koff = (lane >> 4) << 1;                  // 0 or 2
  const float* aptr = rowbase + (size_t)m * IPOT_K + koff;

  v8f acc = {0.f, 0.f, 0.f, 0.f, 0.f, 0.f, 0.f, 0.f};
#pragma unroll 4
  for (int kb = 0; kb < IPOT_K; kb += 4) {
    v2f af = *(const v2f*)(aptr + kb);
    if (EXPMODE) { af.x = __expf(2.0f * af.x); af.y = __expf(2.0f * af.y); }
    v2f bf = *(const v2f*)(&sigma_s[kb + koff]);
    acc = __builtin_amdgcn_wmma_f32_16x16x4_f32(
        /*neg_a=*/false, af, /*neg_b=*/false, bf,
        /*c_mod=*/(short)0, acc, /*reuse_a=*/false, /*reuse_b=*/false);
  }
  // C layout: VGPR v, lanes 0-15 -> row v ; lanes 16-31 -> row 8+v (all cols equal).
  if (lane == 0 || lane == 16) {
    const int rb = (lane >> 4) * 8;
#pragma unroll
    for (int v = 0; v < 8; ++v)
      r_s[wave][rb + v] = (1.0f / (float)IPOT_N) / (acc[v] + IPOT_EPS);
  }
  __syncthreads();

  float rv[ROWS_PER_WAVE];
#pragma unroll
  for (int v = 0; v < ROWS_PER_WAVE; ++v) rv[v] = r_s[wave][v];

  if (FINAL) {
    // out[i][j] = N * r_i * K[i][j] * sigma_j   (coalesced b128 I/O)
    for (int row = 0; row < ROWS_PER_WAVE; ++row) {
      const float s = (float)IPOT_N * rv[row];
      const float* rp = rowbase + (size_t)row * IPOT_K;
      float* op = out + ((size_t)tile * ROWS_PER_WAVE + row) * IPOT_K;
#pragma unroll
      for (int g = 0; g < 4; ++g) {
        const int cb = g * 128 + 4 * lane;
        v4f kv = *(const v4f*)(rp + cb);
        if (EXPMODE) kv = ipot_exp4(kv);
        const v4f sv = *(const v4f*)(&sigma_s[cb]);
        v4f ov;
        ov.x = s * kv.x * sv.x;
        ov.y = s * kv.y * sv.y;
        ov.z = s * kv.z * sv.z;
        ov.w = s * kv.w * sv.w;
        *(v4f*)(op + cb) = ov;
      }
    }
  } else {
    // ---- Phase B: colsum partials. Lane owns cols {4L..4L+3}+128g, g=0..3. ----
    float tacc[16] = {0.f};
    for (int row = 0; row < ROWS_PER_WAVE; ++row) {
      const float rr = rv[row];
      const float* rp = rowbase + (size_t)row * IPOT_K;
#pragma unroll
      for (int g = 0; g < 4; ++g) {
        const int cb = g * 128 + 4 * lane;
        v4f kv = *(const v4f*)(rp + cb);
        if (EXPMODE) kv = ipot_exp4(kv);
        tacc[4 * g + 0] = fmaf(rr, kv.x, tacc[4 * g + 0]);
        tacc[4 * g + 1] = fmaf(rr, kv.y, tacc[4 * g + 1]);
        tacc[4 * g + 2] = fmaf(rr, kv.z, tacc[4 * g + 2]);
        tacc[4 * g + 3] = fmaf(rr, kv.w, tacc[4 * g + 3]);
      }
    }
    // per-wave partial (scaled by sigma); each (wave, col) written by exactly one lane
#pragma unroll
    for (int g = 0; g < 4; ++g) {
      const int cb = g * 128 + 4 * lane;
      const v4f sv = *(const v4f*)(&sigma_s[cb]);
      v4f pv;
      pv.x = tacc[4 * g + 0] * sv.x;
      pv.y = tacc[4 * g + 1] * sv.y;
      pv.z = tacc[4 * g + 2] * sv.z;
      pv.w = tacc[4 * g + 3] * sv.w;
      *(v4f*)(&ps[wave][cb]) = pv;
    }
    __syncthreads();
    // deterministic 4-wave reduce -> one partial row per block
    for (int i = t; i < IPOT_K; i += WG_SIZE) {
      const float s = ps[0][i] + ps[1][i] + ps[2][i] + ps[3][i];
      partial[(size_t)blockIdx.x * IPOT_K + i] = s;
    }
  }
}

// ---------------- sigma update: fixed-order reduction over 512 partials ----
__global__ __launch_bounds__(512)
void ipot_sigma(const float* __restrict__ partial, float* __restrict__ sigma) {
  __shared__ float red[16][32];
  const int t    = threadIdx.x;
  const int colL = t & 31;
  const int seg  = t >> 5;                      // 0..15, 32 partial-rows each
  const int col  = blockIdx.x * 32 + colL;      // 16 blocks x 32 cols
  float s = 0.f;
  const int w0 = seg * 32;
#pragma unroll 8
  for (int w = 0; w < 32; ++w)
    s += partial[(size_t)(w0 + w) * IPOT_K + col];
  red[seg][colL] = s;
  __syncthreads();
  if (seg == 0) {
    float tot = 0.f;
#pragma unroll
    for (int q = 0; q < 16; ++q) tot += red[q][colL];
    sigma[col] = (1.0f / (float)IPOT_K) / (tot + IPOT_EPS);
  }
}

// ---------------------------------------------------------------------------
extern "C" void kernel_launch(void* const* d_in, const int* in_sizes, int n_in,
                              void* d_out, int out_size, void* d_ws, size_t ws_size,
                              hipStream_t stream) {
  (void)in_sizes; (void)n_in; (void)out_size;
  const float* C  = (const float*)d_in[0];
  float* out      = (float*)d_out;
  char* ws        = (char*)d_ws;

  const size_t kmat_bytes    = (size_t)IPOT_N * IPOT_K * sizeof(float);   // 64 MB
  const size_t partial_bytes = (size_t)GRID_MAIN * IPOT_K * sizeof(float);// 1 MB
  const size_t sigma_bytes   = (size_t)IPOT_K * sizeof(float);            // 2 KB
  const bool materialize = ws_size >= kmat_bytes + partial_bytes + sigma_bytes;

  float *Km, *partial, *sigma;
  if (materialize) {
    Km      = (float*)ws;
    partial = (float*)(ws + kmat_bytes);
    sigma   = (float*)(ws + kmat_bytes + partial_bytes);
  } else {
    Km      = nullptr;
    partial = (float*)ws;
    sigma   = (float*)(ws + partial_bytes);
  }

  ipot_seed<<<1, IPOT_K, 0, stream>>>(sigma);
  if (materialize) {
    const int total4 = IPOT_N * IPOT_K / 4;
    ipot_expm<<<(total4 + 255) / 256, 256, 0, stream>>>(C, Km);
  }
  const float* src = materialize ? (const float*)Km : C;

  // iterations 1..19: fused rowsum+colsum pass, then sigma update
  for (int it = 0; it < 19; ++it) {
    if (materialize)
      ipot_pass<false, false><<<GRID_MAIN, WG_SIZE, 0, stream>>>(src, sigma, partial, nullptr);
    else
      ipot_pass<true, false><<<GRID_MAIN, WG_SIZE, 0, stream>>>(src, sigma, partial, nullptr);
    ipot_sigma<<<IPOT_K / 32, 512, 0, stream>>>(partial, sigma);
  }
  // iteration 20: row-normalize with sigma^(19) and emit Q*n (last sigma update unused)
  if (materialize)
    ipot_pass<false, true><<<GRID_MAIN, WG_SIZE, 0, stream>>>(src, sigma, nullptr, out);
  else
    ipot_pass<true, true><<<GRID_MAIN, WG_SIZE, 0, stream>>>(src, sigma, nullptr, out);
}